// Network_13941463843508
// MI455X (gfx1250) — compile-verified
//
#include <hip/hip_runtime.h>

// ---------------------------------------------------------------------------
// SLAYER SRM-alpha 3-layer SNN for MI455X (gfx1250, wave32).
//
// Shapes: X [64,4,2000], W1 [512,4], W2 [2,512] -> out [64,2,2000] (f32).
//
// Strategy:
//  * fc1 (K=4) maps exactly onto V_WMMA_F32_16X16X4_F32: Z1[bt][h] = X.W1^T,
//    written to workspace with 64B-coalesced stores (if ws fits ~250MB).
//  * One workgroup per batch runs the inescapably-sequential T=2000 scan of
//    all three layers in lockstep: layer1 spikes -> shfl_xor wave reduction ->
//    double-buffered LDS partials (ONE barrier/step) -> wave0 carries
//    layer-2/3 state in registers and writes the psp readout.
//  * Fallback (small ws): fc1 fused into the scan (4 FMAs/ch/step, X row in LDS).
// ---------------------------------------------------------------------------

typedef __attribute__((ext_vector_type(2))) float v2f;
typedef __attribute__((ext_vector_type(8))) float v8f;

#define TSTEPS 2000
#define HID    512
#define NINP   4
#define NOUTP  2
#define BATCH  64

// Neuron constants (TAU_S=10, TAU_R=1, TS=1, THETA=10)
#define K_DS 0.90483741803595952f   // exp(-1/10)
#define K_DR 0.36787944117144233f   // exp(-1)
#define K_CS 0.27182818284590452f   // e/10
#define K_CR (-54.365636569180905f) // -2*10*e
#define K_TH 10.0f

__device__ __forceinline__ float wave_reduce_add(float v) {
    v += __shfl_xor(v, 1, 32);
    v += __shfl_xor(v, 2, 32);
    v += __shfl_xor(v, 4, 32);
    v += __shfl_xor(v, 8, 32);
    v += __shfl_xor(v, 16, 32);
    return v;
}

// ---------------------------------------------------------------------------
// Kernel 1: Z1[bt][h] = sum_k X[b,k,t] * W1[h,k]   via V_WMMA_F32_16X16X4_F32
// D(16x16) = A(16x4: M=bt tile of X) x B(4x16: N=h tile of W1).
// One wave owns one bt-tile and sweeps all 32 h-tiles (A reused, W1 L2-hot).
// ---------------------------------------------------------------------------
__global__ __launch_bounds__(256) void fc1_wmma(const float* __restrict__ X,
                                                const float* __restrict__ W1,
                                                float* __restrict__ Z1) {
    const int lane   = threadIdx.x & 31;
    const int wave   = threadIdx.x >> 5;
    const int btTile = blockIdx.x * 8 + wave;       // 8000 tiles total
    const int n0     = btTile * 16;                 // base bt column
    const int half   = lane >> 4;                   // 0: K={0,1}, 1: K={2,3}
    const int idx16  = lane & 15;

    // A tile: A[m][k] = X[b, k, t], m = bt row (lane&15), K pair by half-wave.
    const int col   = n0 + idx16;
    const int bb    = col / TSTEPS;
    const int tt    = col - bb * TSTEPS;
    const int kbase = half * 2;
    const float* Xb = X + (size_t)bb * NINP * TSTEPS;
    v2f A;
    A.x = Xb[(size_t)kbase * TSTEPS + tt];
    A.y = Xb[(size_t)(kbase + 1) * TSTEPS + tt];

    for (int hT = 0; hT < HID / 16; ++hT) {
        // B tile: B[k][n] = W1[h, k]; rows are 16B so this is a packed pair.
        const int h = hT * 16 + idx16;
        v2f Bv;
        Bv.x = W1[h * NINP + kbase];
        Bv.y = W1[h * NINP + kbase + 1];

        v8f C = {0.f, 0.f, 0.f, 0.f, 0.f, 0.f, 0.f, 0.f};
        v8f D = __builtin_amdgcn_wmma_f32_16x16x4_f32(
            /*neg_a=*/false, A, /*neg_b=*/false, Bv,
            /*c_mod=*/(short)0, C, /*reuse_a=*/false, /*reuse_b=*/false);

        // D layout: VGPR j -> M = j + 8*half, N = idx16.
        // Z1 row-major [bt][HID]: each half-wave stores 16 contiguous dwords.
        float* dst = Z1 + (size_t)n0 * HID + hT * 16 + idx16;
        #pragma unroll
        for (int j = 0; j < 8; ++j) {
            const int m = j + 8 * half;
            dst[(size_t)m * HID] = D[j];
        }
    }
}

// ---------------------------------------------------------------------------
// Kernel 2: fused 3-layer temporal scan. One block per batch, 256 threads,
// 2 hidden channels per thread. FUSED=true computes z1 on the fly (no ws).
// ---------------------------------------------------------------------------
template <bool FUSED>
__global__ __launch_bounds__(256) void slayer_scan(const float* __restrict__ X,
                                                   const float* __restrict__ W1,
                                                   const float* __restrict__ W2,
                                                   const float* __restrict__ Z1,
                                                   float* __restrict__ out) {
    const int b    = blockIdx.x;
    const int tid  = threadIdx.x;
    const int lane = tid & 31;
    const int wave = tid >> 5;   // 8 waves

    __shared__ float part[2][2][8];                    // [buf][o][wave]
    __shared__ float xsh[FUSED ? NINP * TSTEPS : 1];   // staged X row (32 KB)

    const int hA = tid;
    const int hB = tid + 256;

    float w1A[NINP] = {0.f, 0.f, 0.f, 0.f};
    float w1B[NINP] = {0.f, 0.f, 0.f, 0.f};
    if constexpr (FUSED) {
        for (int i = tid; i < NINP * TSTEPS; i += 256)
            xsh[i] = X[(size_t)b * NINP * TSTEPS + i];
        #pragma unroll
        for (int i = 0; i < NINP; ++i) {
            w1A[i] = W1[hA * NINP + i];
            w1B[i] = W1[hB * NINP + i];
        }
        __syncthreads();
    }

    // Layer-1 SRM state (2 channels / thread)
    float xsA = 0.f, ysA = 0.f, xrA = 0.f, yrA = 0.f;
    float xsB = 0.f, ysB = 0.f, xrB = 0.f, yrB = 0.f;
    const float w2A0 = W2[hA],       w2A1 = W2[HID + hA];
    const float w2B0 = W2[hB],       w2B1 = W2[HID + hB];

    // Layer-2/3 state: carried redundantly by every lane of wave 0.
    float xs20 = 0.f, ys20 = 0.f, xr20 = 0.f, yr20 = 0.f;
    float xs21 = 0.f, ys21 = 0.f, xr21 = 0.f, yr21 = 0.f;
    float xs30 = 0.f, ys30 = 0.f, xs31 = 0.f, ys31 = 0.f;

    const float* z1b = FUSED ? nullptr : (Z1 + (size_t)b * TSTEPS * HID);
    float* outb = out + (size_t)b * NOUTP * TSTEPS;

    for (int t = 0; t < TSTEPS; ++t) {
        // ---- layer 1: spike from pre-update state ----
        const float sA = (K_CS * ysA + K_CR * yrA >= K_TH) ? 1.0f : 0.0f;
        const float sB = (K_CS * ysB + K_CR * yrB >= K_TH) ? 1.0f : 0.0f;

        float zA, zB;
        if constexpr (FUSED) {
            const float x0 = xsh[t];
            const float x1 = xsh[TSTEPS + t];
            const float x2 = xsh[2 * TSTEPS + t];
            const float x3 = xsh[3 * TSTEPS + t];
            zA = w1A[0] * x0 + w1A[1] * x1 + w1A[2] * x2 + w1A[3] * x3;
            zB = w1B[0] * x0 + w1B[1] * x1 + w1B[2] * x2 + w1B[3] * x3;
        } else {
            zA = z1b[(size_t)t * HID + hA];
            zB = z1b[(size_t)t * HID + hB];
            if (t + 8 < TSTEPS)  // gfx1250 global_prefetch_b8 for upcoming rows
                __builtin_prefetch(&z1b[(size_t)(t + 8) * HID + hA], 0, 0);
        }

        // ---- layer 1: state update (alpha PSP + refractory, TS=1) ----
        xsA = K_DS * xsA + zA;  ysA = K_DS * (ysA + xsA);
        xrA = K_DR * xrA + sA;  yrA = K_DR * (yrA + xrA);
        xsB = K_DS * xsB + zB;  ysB = K_DS * (ysB + xsB);
        xrB = K_DR * xrB + sB;  yrB = K_DR * (yrB + xrB);

        // ---- fc2 partial: z2[o] = sum_h W2[o,h] * s1[h] ----
        float p0 = sA * w2A0 + sB * w2B0;
        float p1 = sA * w2A1 + sB * w2B1;
        p0 = wave_reduce_add(p0);
        p1 = wave_reduce_add(p1);
        const int buf = t & 1;  // double buffer => single barrier per step
        if (lane == 0) { part[buf][0][wave] = p0; part[buf][1][wave] = p1; }
        __syncthreads();

        if (wave == 0) {
            // lanes 0-7 reduce output 0, lanes 8-15 output 1 (16-31 mirror)
            const int sel = (lane >> 3) & 1;
            float v = part[buf][sel][lane & 7];
            v += __shfl_xor(v, 1, 32);
            v += __shfl_xor(v, 2, 32);
            v += __shfl_xor(v, 4, 32);
            const float z20 = __shfl(v, 0, 32);
            const float z21 = __shfl(v, 8, 32);

            // layer 2 spikes + layer 3 psp readout (pre-update states)
            const float s20 = (K_CS * ys20 + K_CR * yr20 >= K_TH) ? 1.0f : 0.0f;
            const float s21 = (K_CS * ys21 + K_CR * yr21 >= K_TH) ? 1.0f : 0.0f;
            const float o0 = K_CS * ys30;
            const float o1 = K_CS * ys31;

            // layer 2 state update
            xs20 = K_DS * xs20 + z20;  ys20 = K_DS * (ys20 + xs20);
            xr20 = K_DR * xr20 + s20;  yr20 = K_DR * (yr20 + xr20);
            xs21 = K_DS * xs21 + z21;  ys21 = K_DS * (ys21 + xs21);
            xr21 = K_DR * xr21 + s21;  yr21 = K_DR * (yr21 + xr21);

            // layer 3 (pure psp) state update driven by layer-2 spikes
            xs30 = K_DS * xs30 + s20;  ys30 = K_DS * (ys30 + xs30);
            xs31 = K_DS * xs31 + s21;  ys31 = K_DS * (ys31 + xs31);

            if (lane == 0) {
                outb[t] = o0;
                outb[TSTEPS + t] = o1;
            }
        }
        // No second barrier: parity-buffered `part` + the next iteration's
        // barrier order the t+2 overwrite after wave0's read at t.
    }
}

// ---------------------------------------------------------------------------
extern "C" void kernel_launch(void* const* d_in, const int* in_sizes, int n_in,
                              void* d_out, int out_size, void* d_ws, size_t ws_size,
                              hipStream_t stream) {
    const float* X  = (const float*)d_in[0];  // [64,4,2000]
    const float* W1 = (const float*)d_in[1];  // [512,4]
    const float* W2 = (const float*)d_in[2];  // [2,512]
    float* out = (float*)d_out;               // [64,2,2000]

    const size_t Z1_BYTES = (size_t)BATCH * TSTEPS * HID * sizeof(float); // 250 MB

    if (ws_size >= Z1_BYTES) {
        float* Z1 = (float*)d_ws;
        // 128000 bt columns / 16 per tile / 8 waves per block = 1000 blocks
        fc1_wmma<<<(BATCH * TSTEPS) / (16 * 8), 256, 0, stream>>>(X, W1, Z1);
        slayer_scan<false><<<BATCH, 256, 0, stream>>>(X, W1, W2, Z1, out);
    } else {
        slayer_scan<true><<<BATCH, 256, 0, stream>>>(X, W1, W2, (const float*)nullptr, out);
    }
}